// QuantumCircuitLayer_31619549233444
// MI455X (gfx1250) — compile-verified
//
#include <hip/hip_runtime.h>
#include <math.h>

// CDNA5 (gfx1250) wave32 WMMA formulation of the quantum-circuit layer.
// The circuit after the initial cos/sin is an exact linear map u = T v with a
// batch-independent 12x12 f32 matrix T; we apply it with V_WMMA_F32_16X16X4_F32
// (3 chained K=4 steps), 16 batch rows per wave tile. Memory-bound: ~201 MB of
// traffic -> ~8.6 us floor at 23.3 TB/s HBM.

typedef __attribute__((ext_vector_type(2))) float v2f;
typedef __attribute__((ext_vector_type(8))) float v8f;

#define N_Q 6

// ---------------------------------------------------------------------------
// Prelude: single thread composes the 12x12 transform T = M_L ... M_1 and
// stores it into d_ws in the WMMA B-matrix layout: Bm[k*16 + n] = T[n][k]
// (n = output index 0..11, padded with zeros to n<16; k = input index 0..11).
// ---------------------------------------------------------------------------
__global__ void qc_prelude(const float* __restrict__ rp,   // (L,6,3)
                           const float* __restrict__ ep,   // (L,5)
                           float* __restrict__ Bm,         // 12*16 floats
                           int nLayers) {
    if (threadIdx.x != 0 || blockIdx.x != 0) return;

    float T[12][12];
    for (int i = 0; i < 12; ++i)
        for (int j = 0; j < 12; ++j)
            T[i][j] = (i == j) ? 1.0f : 0.0f;

    for (int layer = 0; layer < nLayers; ++layer) {
        float c[N_Q], s[N_Q];
        for (int q = 0; q < N_Q; ++q) {
            float rx = rp[(layer * N_Q + q) * 3 + 0];
            c[q] = cosf(rx * 0.5f);
            s[q] = sinf(rx * 0.5f);
        }
        float st[N_Q - 1];
        for (int i = 0; i < N_Q - 1; ++i)
            st[i] = 1.0f / (1.0f + expf(-ep[layer * (N_Q - 1) + i]));

        // Entanglement matrix E (6x6): left[i] = (1-st/2)*r_i + (st/2)*r_{i+1},
        // last = (1-st4/2)*r_5 + (st4/2)*r_4.
        float E[N_Q][N_Q];
        for (int i = 0; i < N_Q; ++i)
            for (int j = 0; j < N_Q; ++j) E[i][j] = 0.0f;
        for (int i = 0; i < N_Q - 1; ++i) {
            E[i][i]     = 1.0f - 0.5f * st[i];
            E[i][i + 1] = 0.5f * st[i];
        }
        E[N_Q - 1][N_Q - 1] = 1.0f - 0.5f * st[N_Q - 2];
        E[N_Q - 1][N_Q - 2] = 0.5f * st[N_Q - 2];

        // Layer matrix M (12x12): real' = E(C r + S i), imag' = C i - S r.
        float M[12][12];
        for (int i = 0; i < 12; ++i)
            for (int j = 0; j < 12; ++j) M[i][j] = 0.0f;
        for (int i = 0; i < N_Q; ++i)
            for (int j = 0; j < N_Q; ++j) {
                M[i][j]         = E[i][j] * c[j];
                M[i][j + N_Q]   = E[i][j] * s[j];
            }
        for (int i = 0; i < N_Q; ++i) {
            M[i + N_Q][i]       = -s[i];
            M[i + N_Q][i + N_Q] =  c[i];
        }

        // T = M * T
        float Tn[12][12];
        for (int i = 0; i < 12; ++i)
            for (int j = 0; j < 12; ++j) {
                float acc = 0.0f;
                for (int k = 0; k < 12; ++k) acc += M[i][k] * T[k][j];
                Tn[i][j] = acc;
            }
        for (int i = 0; i < 12; ++i)
            for (int j = 0; j < 12; ++j) T[i][j] = Tn[i][j];
    }

    // B layout: B[k][n] = T[n][k], zero-padded to N=16.
    for (int k = 0; k < 12; ++k)
        for (int n = 0; n < 16; ++n)
            Bm[k * 16 + n] = (n < 12) ? T[n][k] : 0.0f;
}

// ---------------------------------------------------------------------------
// Main kernel: one 16-row tile per wave per iteration.
//   A (16x4 f32, per K-chunk): lanes 0-15 hold {K=0,K=1} of row M=lane,
//                              lanes 16-31 hold {K=2,K=3} of row M=lane-16.
//   B (4x16 f32): lanes 0-15 hold {K=0,K=1} of column N=lane,
//                 lanes 16-31 hold {K=2,K=3} of column N=lane-16.
//   D (16x16 f32): VGPR j, lanes 0-15 -> M=j, lanes 16-31 -> M=8+j.
// ---------------------------------------------------------------------------
__global__ void __launch_bounds__(256)
qc_main(const float* __restrict__ x, const float* __restrict__ Bm,
        float* __restrict__ out, int nTiles) {
    __shared__ float smem[8 * 16 * 20];   // 8 waves * 16x16 tile (stride 20)

    const int tid        = threadIdx.x;
    const int lane       = tid & 31;
    const int waveInBlk  = tid >> 5;
    const int waveId     = blockIdx.x * (blockDim.x >> 5) + waveInBlk;
    const int totalWaves = gridDim.x * (blockDim.x >> 5);
    float* lds = smem + waveInBlk * (16 * 20);

    const int  n    = lane & 15;
    const bool hi   = lane >= 16;
    const int  koff = hi ? 2 : 0;

    // Load composite matrix in B layout (once; L2-resident, 768 B).
    v2f b0, b1, b2;
    b0.x = Bm[(0 + koff) * 16 + n];
    b0.y = Bm[(1 + koff) * 16 + n];
    b1.x = Bm[(4 + koff) * 16 + n];
    b1.y = Bm[(5 + koff) * 16 + n];
    b2.x = Bm[(8 + koff) * 16 + n];
    b2.y = Bm[(9 + koff) * 16 + n];

    const float PI2 = 1.57079632679489662f;   // pi/2

    for (int tile = waveId; tile < nTiles; tile += totalWaves) {
        // Lane pair (n, n+16) shares row tile*16+n: 3x global_load_b64 each.
        const float* xp = x + (size_t)(tile * 16 + n) * N_Q;
        const float2 p0 = reinterpret_cast<const float2*>(xp)[0];
        const float2 p1 = reinterpret_cast<const float2*>(xp)[1];
        const float2 p2 = reinterpret_cast<const float2*>(xp)[2];
        const float h0 = p0.x * PI2, h1 = p0.y * PI2;
        const float h2 = p1.x * PI2, h3 = p1.y * PI2;
        const float h4 = p2.x * PI2, h5 = p2.y * PI2;

        // v[k] = cos(h_k) (k<6), sin(h_{k-6}) (k>=6); sin z = cos(z - pi/2).
        // chunk0 lanes lo:{v0,v1} hi:{v2,v3}
        // chunk1 lanes lo:{v4,v5} hi:{v6,v7}
        // chunk2 lanes lo:{v8,v9} hi:{v10,v11}
        v2f a0, a1, a2;
        a0.x = __cosf(hi ? h2 : h0);
        a0.y = __cosf(hi ? h3 : h1);
        a1.x = __cosf(hi ? (h0 - PI2) : h4);
        a1.y = __cosf(hi ? (h1 - PI2) : h5);
        a2.x = __cosf((hi ? h4 : h2) - PI2);
        a2.y = __cosf((hi ? h5 : h3) - PI2);

        // D = A0*B0 + A1*B1 + A2*B2  (u_n for 16 rows, f32 matrix cores)
        v8f acc = {};
        acc = __builtin_amdgcn_wmma_f32_16x16x4_f32(false, a0, false, b0,
                                                    (short)0, acc, false, false);
        acc = __builtin_amdgcn_wmma_f32_16x16x4_f32(false, a1, false, b1,
                                                    (short)0, acc, false, false);
        acc = __builtin_amdgcn_wmma_f32_16x16x4_f32(false, a2, false, b2,
                                                    (short)0, acc, false, false);

        // Spill D through per-wave LDS (stride 20 dwords: lanes 0-15 and
        // 16-31 land in disjoint bank groups per ds_store).
        const int mbase = hi ? 8 : 0;
#pragma unroll
        for (int j = 0; j < 8; ++j)
            lds[(mbase + j) * 20 + n] = acc[j];

        // In-wave RAW on LDS: explicit split-counter wait (CDNA5).
        asm volatile("s_wait_dscnt 0" ::: "memory");

        // out[row][q] = sqrt(u_q^2 + u_{q+6}^2); 96 outputs = 3 per lane,
        // fully coalesced 4B stores. Operands are in [0,2] (state magnitude
        // <= sqrt(2)), so the raw v_sqrt_f32 (~1 ulp) is sufficient and
        // avoids the correctly-rounded sqrt's 15-op fixup tail.
        float* outp = out + (size_t)tile * 96;
#pragma unroll
        for (int i = 0; i < 3; ++i) {
            const int e = lane + i * 32;          // 0..95
            const int r = e / 6;
            const int q = e - r * 6;
            const float re = lds[r * 20 + q];
            const float im = lds[r * 20 + q + 6];
            outp[e] = __builtin_amdgcn_sqrtf(re * re + im * im);
        }
        // Next iteration's ds_stores are program-ordered after these ds_loads
        // (LDS ops from one wave execute in order), so no extra barrier needed.
    }
}

extern "C" void kernel_launch(void* const* d_in, const int* in_sizes, int n_in,
                              void* d_out, int out_size, void* d_ws, size_t ws_size,
                              hipStream_t stream) {
    const float* x  = (const float*)d_in[0];
    const float* rp = (const float*)d_in[1];
    const float* ep = (const float*)d_in[2];
    float* out = (float*)d_out;
    float* Bm  = (float*)d_ws;          // 12*16 floats = 768 B

    const int rows    = in_sizes[0] / N_Q;          // 4194304
    const int nLayers = in_sizes[1] / (N_Q * 3);    // 3
    const int nTiles  = rows / 16;                  // 262144 (rows % 16 == 0)

    qc_prelude<<<1, 64, 0, stream>>>(rp, ep, Bm, nLayers);

    // 2048 blocks x 256 threads = 16384 waves -> exactly 16 tiles per wave,
    // wave-uniform loop so EXEC is all-ones at every WMMA.
    qc_main<<<2048, 256, 0, stream>>>(x, Bm, out, nTiles);
}